// PatchLoss_5222680232118
// MI455X (gfx1250) — compile-verified
//
#include <hip/hip_runtime.h>
#include <math.h>

typedef __attribute__((ext_vector_type(2))) float v2f;
typedef __attribute__((ext_vector_type(8))) float v8f;

#define Lc 4
#define Bc 8
#define Nsc 1200
#define Nlc 128
#define Cc 768
#define Hc 448
#define Wc 448
#define GS 56   /* H/8  */
#define GL 16   /* H/28 */
#define MS (GS*GS)  /* 3136 */
#define ML (GL*GL)  /* 256  */

// ---------------------------------------------------------------------------
// Kernel 1: validity grids from f_k (strided sample > 0)
// ---------------------------------------------------------------------------
__global__ void pl_valid_kernel(const float* __restrict__ fk,
                                int* __restrict__ valid_s,
                                int* __restrict__ valid_l) {
  int t = blockIdx.x * blockDim.x + threadIdx.x;
  const int ns = Bc * MS;
  if (t < ns) {
    int b = t / MS, r = t % MS;
    int y = r / GS, x = r % GS;
    valid_s[t] = (fk[(size_t)b * Hc * Wc + (size_t)(y * 8) * Wc + x * 8] > 0.f) ? 1 : 0;
  } else {
    int u = t - ns;
    if (u < Bc * ML) {
      int b = u / ML, r = u % ML;
      int y = r / GL, x = r % GL;
      valid_l[u] = (fk[(size_t)b * Hc * Wc + (size_t)(y * 28) * Wc + x * 28] > 0.f) ? 1 : 0;
    }
  }
}

// ---------------------------------------------------------------------------
// Kernel 2: streaming WMMA kernel.
// One wave per 16-token tile of one (l,b). For each K-chunk of 4 channels,
// build diff = anchor - sel in the A layout of V_WMMA_F32_16X16X4_F32
// (lane m<16: row m, K = k0,k0+1 ; lane 16+m: row m, K = k0+2,k0+3).
// This register pair doubles as the matching B (4x16) tile for diff^T, so
//   acc = wmma_f32_16x16x4_f32(a, a, acc)
// accumulates the Gram matrix; its diagonal is ||anchor - sel_n||^2.
// Simultaneously accumulates sum(|sel|) per token for col_valid.
// ---------------------------------------------------------------------------
__global__ __launch_bounds__(256)
void pl_dist_abs_kernel(const float* __restrict__ sel,
                        const float* __restrict__ anchor,
                        float* __restrict__ dist,
                        float* __restrict__ abss,
                        int N) {
  __shared__ float sh[8 * 288];  // per-wave: 8x32 Gram slab + 32 abs partials
  const int wave = threadIdx.x >> 5;
  const int lane = threadIdx.x & 31;
  const int gw = blockIdx.x * (blockDim.x >> 5) + wave;
  const int tilesPerLB = N >> 4;
  const int tile = gw % tilesPerLB;
  const int lb = gw / tilesPerLB;        // l*B + b, 0..31
  const int half = lane >> 4;            // K sub-pair select
  const int m = lane & 15;               // token row within tile

  const float* __restrict__ srow = sel + ((size_t)lb * N + (size_t)tile * 16 + m) * Cc;
  const float* __restrict__ arow = anchor + (size_t)lb * Cc;

  v8f acc = {};
  float asum = 0.f;
#pragma unroll 4
  for (int kc = 0; kc < Cc; kc += 4) {
    const int k0 = kc + half * 2;
    if ((kc & 63) == 0 && kc + 64 < Cc)
      __builtin_prefetch(srow + kc + 64, 0, 1);   // global_prefetch_b8
    float s0 = srow[k0];
    float s1 = srow[k0 + 1];
    asum += __builtin_fabsf(s0) + __builtin_fabsf(s1);
    v2f a;
    a[0] = arow[k0] - s0;
    a[1] = arow[k0 + 1] - s1;
    // D = A * A(^T-layout) + C : diagonal = squared distances
    acc = __builtin_amdgcn_wmma_f32_16x16x4_f32(false, a, false, a,
                                                (short)0, acc, false, false);
  }

  // Spill Gram slab + abs partials to LDS, pull out the diagonal.
  float* shc = &sh[wave * 288];
#pragma unroll
  for (int r = 0; r < 8; ++r) shc[r * 32 + lane] = acc[r];
  shc[256 + lane] = asum;
  __syncthreads();

  if (lane < 16) {
    const int i = lane;  // token index in tile
    // C/D layout: VGPR r holds M=r (lanes 0-15) and M=r+8 (lanes 16-31, N=lane-16)
    float d2 = (i < 8) ? shc[i * 32 + i] : shc[(i - 8) * 32 + (i + 8)];
    float at = shc[256 + i] + shc[256 + i + 16];  // k%4 in {0,1} half + {2,3} half
    size_t o = (size_t)lb * N + (size_t)tile * 16 + i;
    dist[o] = sqrtf(d2 > 0.f ? d2 : 0.f);
    abss[o] = at;
  }
}

// ---------------------------------------------------------------------------
// Kernel 3: col_valid[l,n] = (sum_b abssum[l,b,n]) != 0
// ---------------------------------------------------------------------------
__global__ void pl_colvalid_kernel(const float* __restrict__ abss,
                                   int* __restrict__ cv, int N) {
  int t = blockIdx.x * blockDim.x + threadIdx.x;
  if (t >= Lc * N) return;
  int l = t / N, n = t % N;
  float s = 0.f;
#pragma unroll
  for (int b = 0; b < Bc; ++b) s += abss[((size_t)(l * Bc + b)) * N + n];
  cv[t] = (s != 0.f) ? 1 : 0;
}

// ---------------------------------------------------------------------------
// Kernel 4: per-(l,b) masked means + softplus for both scales.
// ---------------------------------------------------------------------------
__global__ __launch_bounds__(256)
void pl_stats_kernel(const float* __restrict__ dist_s, const int* __restrict__ idx_s,
                     const int* __restrict__ valid_s, const int* __restrict__ cv_s,
                     const float* __restrict__ dist_l, const int* __restrict__ idx_l,
                     const int* __restrict__ valid_l, const int* __restrict__ cv_l,
                     float* __restrict__ lo_s, int* __restrict__ ac_s,
                     float* __restrict__ lo_l, int* __restrict__ ac_l) {
  __shared__ float shp[256], shn[256];
  __shared__ int shcp[256], shcn[256];
  const int lb = blockIdx.x;
  const int l = lb / Bc, b = lb % Bc;
  const int tid = threadIdx.x;

  for (int sc = 0; sc < 2; ++sc) {
    const float* dist = sc ? dist_l : dist_s;
    const int* idx   = sc ? idx_l   : idx_s;
    const int* valid = sc ? valid_l : valid_s;
    const int* cv    = sc ? cv_l    : cv_s;
    const int N = sc ? Nlc : Nsc;
    const int M = sc ? ML  : MS;

    float sp = 0.f, sn = 0.f;
    int cp = 0, cn = 0;
    for (int n = tid; n < N; n += blockDim.x) {
      if (cv[l * N + n]) {
        int member = valid[b * M + idx[(size_t)lb * N + n]];
        float d = dist[(size_t)lb * N + n];
        if (member) { sp += d; ++cp; } else { sn += d; ++cn; }
      }
    }
    shp[tid] = sp; shn[tid] = sn; shcp[tid] = cp; shcn[tid] = cn;
    __syncthreads();
    for (int s = 128; s > 0; s >>= 1) {
      if (tid < s) {
        shp[tid] += shp[tid + s];  shn[tid] += shn[tid + s];
        shcp[tid] += shcp[tid + s]; shcn[tid] += shcn[tid + s];
      }
      __syncthreads();
    }
    if (tid == 0) {
      int pc = shcp[0], nc = shcn[0];
      float ap = shp[0] / (float)(pc > 1 ? pc : 1);
      float an = shn[0] / (float)(nc > 1 ? nc : 1);
      float x = ap - an;
      // numerically stable softplus = log(1 + exp(x))
      float lossv = (x > 0.f) ? x + log1pf(expf(-x)) : log1pf(expf(x));
      if (sc == 0) { lo_s[lb] = lossv; ac_s[lb] = (pc > 0 && nc > 0) ? 1 : 0; }
      else         { lo_l[lb] = lossv; ac_l[lb] = (pc > 0 && nc > 0) ? 1 : 0; }
    }
    __syncthreads();
  }
}

// ---------------------------------------------------------------------------
// Kernel 5: final scalar combine (64 entries -> 1 float)
// ---------------------------------------------------------------------------
__global__ void pl_final_kernel(const float* __restrict__ lo_s, const int* __restrict__ ac_s,
                                const float* __restrict__ lo_l, const int* __restrict__ ac_l,
                                float* __restrict__ out) {
  if (threadIdx.x == 0 && blockIdx.x == 0) {
    float total = 0.f;
    int times = 0;
    for (int i = 0; i < Lc * Bc; ++i) {
      if (ac_s[i]) { total += lo_s[i]; ++times; }
      if (ac_l[i] && ac_s[i]) { total += lo_l[i]; ++times; }
    }
    out[0] = (times > 0) ? total / (float)times : 0.f;
  }
}

// ---------------------------------------------------------------------------
extern "C" void kernel_launch(void* const* d_in, const int* in_sizes, int n_in,
                              void* d_out, int out_size, void* d_ws, size_t ws_size,
                              hipStream_t stream) {
  const float* sel_s = (const float*)d_in[0];
  const int*   idx_s = (const int*)d_in[1];
  const float* sel_l = (const float*)d_in[2];
  const int*   idx_l = (const int*)d_in[3];
  const float* ker_s = (const float*)d_in[4];  // (L,B,C,1) contiguous -> (L,B,C)
  const float* ker_l = (const float*)d_in[5];
  const float* f_k   = (const float*)d_in[6];
  float* out = (float*)d_out;

  char* w = (char*)d_ws;
  float* dist_s = (float*)w;  w += (size_t)Lc * Bc * Nsc * 4;
  float* abs_s  = (float*)w;  w += (size_t)Lc * Bc * Nsc * 4;
  float* dist_l = (float*)w;  w += (size_t)Lc * Bc * Nlc * 4;
  float* abs_l  = (float*)w;  w += (size_t)Lc * Bc * Nlc * 4;
  int*   valid_s = (int*)w;   w += (size_t)Bc * MS * 4;
  int*   valid_l = (int*)w;   w += (size_t)Bc * ML * 4;
  int*   cv_s    = (int*)w;   w += (size_t)Lc * Nsc * 4;
  int*   cv_l    = (int*)w;   w += (size_t)Lc * Nlc * 4;
  float* lo_s    = (float*)w; w += 32 * 4;
  float* lo_l    = (float*)w; w += 32 * 4;
  int*   ac_s    = (int*)w;   w += 32 * 4;
  int*   ac_l    = (int*)w;   w += 32 * 4;

  // 1) validity grids: B*(3136+256) = 27136 threads
  pl_valid_kernel<<<106, 256, 0, stream>>>(f_k, valid_s, valid_l);

  // 2) streaming WMMA: waves = L*B*(N/16); 8 waves per 256-thread block
  pl_dist_abs_kernel<<<(Lc * Bc * (Nsc / 16)) / 8, 256, 0, stream>>>(
      sel_s, ker_s, dist_s, abs_s, Nsc);
  pl_dist_abs_kernel<<<(Lc * Bc * (Nlc / 16)) / 8, 256, 0, stream>>>(
      sel_l, ker_l, dist_l, abs_l, Nlc);

  // 3) col_valid reductions
  pl_colvalid_kernel<<<(Lc * Nsc + 255) / 256, 256, 0, stream>>>(abs_s, cv_s, Nsc);
  pl_colvalid_kernel<<<(Lc * Nlc + 255) / 256, 256, 0, stream>>>(abs_l, cv_l, Nlc);

  // 4) per-(l,b) masked means + softplus
  pl_stats_kernel<<<Lc * Bc, 256, 0, stream>>>(dist_s, idx_s, valid_s, cv_s,
                                               dist_l, idx_l, valid_l, cv_l,
                                               lo_s, ac_s, lo_l, ac_l);

  // 5) scalar combine
  pl_final_kernel<<<1, 32, 0, stream>>>(lo_s, ac_s, lo_l, ac_l, out);
}